// PandGNN_53266184405345
// MI455X (gfx1250) — compile-verified
//
#include <hip/hip_runtime.h>
#include <hip/hip_bf16.h>
#include <math.h>

// Problem constants (match reference)
#define MM 100000
#define NN 50000
#define NNODES 150000            // M + N
#define DD 128
#define EE 2000000
#define BB 8192
#define KK 32
#define TAU 0.8f
#define REGC 1e-4f

#define NB_ELEMS ((size_t)NNODES * DD)      // floats per node buffer
#define NB_VEC4  (NB_ELEMS / 4)             // float4 elements
#define PREF_EDGE_LOOKAHEAD 8192            // edges of L2 prefetch lead

typedef float v2f __attribute__((ext_vector_type(2)));
typedef float v8f __attribute__((ext_vector_type(8)));

// ---------------- helpers ----------------

__device__ __forceinline__ float wave_sum(float v) {
    // full wave32 butterfly reduction; result in all lanes
    #pragma unroll
    for (int off = 16; off >= 1; off >>= 1)
        v += __shfl_xor(v, off, 32);
    return v;
}

__device__ __forceinline__ float dot4(float4 a, float4 b) {
    return a.x * b.x + a.y * b.y + a.z * b.z + a.w * b.w;
}

__device__ __forceinline__ float log_sigmoid(float x) {
    // stable: min(x,0) - log1p(exp(-|x|))
    return fminf(x, 0.f) - log1pf(expf(-fabsf(x)));
}

// read one float4 from the virtual concat [E_user ; E_item] node matrix
__device__ __forceinline__ float4 concat_ld4(const float4* Eu, const float4* Ei,
                                             int row, int j /*0..31*/) {
    return (row < MM) ? Eu[(size_t)row * 32 + j] : Ei[(size_t)(row - MM) * 32 + j];
}

// ---------------- kernels ----------------

__global__ void k_zero(float* out) { out[0] = 0.f; }

// dst <- concat(Eu, Ei)   (one float4 per thread)
__global__ void k_concat_copy(float4* dst, const float4* Eu, const float4* Ei) {
    size_t i = (size_t)blockIdx.x * blockDim.x + threadIdx.x;
    if (i >= NB_VEC4) return;
    int row = (int)(i >> 5);
    int j = (int)(i & 31);
    dst[i] = concat_ld4(Eu, Ei, row, j);
}

__global__ void k_copy(float4* dst, const float4* src) {
    size_t i = (size_t)blockIdx.x * blockDim.x + threadIdx.x;
    if (i >= NB_VEC4) return;
    dst[i] = src[i];
}

// scatter-add with concat source: y[dst] += x0[src]  (2 edges per 256-thr block)
__global__ void k_scatter_concat(float* __restrict__ y, const int* __restrict__ ei,
                                 const float* __restrict__ Eu, const float* __restrict__ Ei) {
    int e = blockIdx.x * 2 + (threadIdx.x >> 7);
    int d = threadIdx.x & 127;
    int src = ei[e];
    int dst = ei[EE + e];
    // warm L2 ahead of the gather stream (global_prefetch)
    int e2 = e + PREF_EDGE_LOOKAHEAD;
    if (e2 < EE) {
        int s2 = ei[e2];
        const float* p = (s2 < MM) ? &Eu[(size_t)s2 * DD + d]
                                   : &Ei[(size_t)(s2 - MM) * DD + d];
        __builtin_prefetch(p, 0, 1);
    }
    float v = (src < MM) ? Eu[(size_t)src * DD + d]
                         : Ei[(size_t)(src - MM) * DD + d];
    atomicAdd(&y[(size_t)dst * DD + d], v);
}

// scatter-add, plain source: y[dst] += x[src]
__global__ void k_scatter(float* __restrict__ y, const int* __restrict__ ei,
                          const float* __restrict__ x) {
    int e = blockIdx.x * 2 + (threadIdx.x >> 7);
    int d = threadIdx.x & 127;
    int src = ei[e];
    int dst = ei[EE + e];
    int e2 = e + PREF_EDGE_LOOKAHEAD;
    if (e2 < EE) {
        int s2 = ei[e2];
        __builtin_prefetch(&x[(size_t)s2 * DD + d], 0, 1);
    }
    atomicAdd(&y[(size_t)dst * DD + d], x[(size_t)src * DD + d]);
}

// x1 <- (x0 + x1 + x2) / 3   (z finalization, in place in x1)
__global__ void k_finalize(float4* x1, const float4* x2,
                           const float4* Eu, const float4* Ei) {
    size_t i = (size_t)blockIdx.x * blockDim.x + threadIdx.x;
    if (i >= NB_VEC4) return;
    int row = (int)(i >> 5);
    int j = (int)(i & 31);
    float4 a = concat_ld4(Eu, Ei, row, j);
    float4 b = x1[i];
    float4 c = x2[i];
    const float s = 1.f / 3.f;
    float4 r;
    r.x = (a.x + b.x + c.x) * s;
    r.y = (a.y + b.y + c.y) * s;
    r.z = (a.z + b.z + c.z) * s;
    r.w = (a.w + b.w + c.w) * s;
    x1[i] = r;
}

// Signed-BPR + reg term.  One wave per triplet b; neg dots via f32 WMMA.
// mode 0: arg = (2 - sgn(w))*pos - neg        (positive graph)
// mode 1: arg = neg - (2 + sgn(w))*pos        (negative graph)
// Contribution to output: -sum(log_sigmoid(arg)) + REG * (|u|^2+|v|^2+sum_k |n_k|^2)
__global__ void k_bpr(const float* __restrict__ z,
                      const int* __restrict__ u, const int* __restrict__ v,
                      const int* __restrict__ n, const float* __restrict__ w,
                      float* out, int mode) {
    __shared__ float su[8][DD];                // staged u-row per wave
    const int wid  = threadIdx.x >> 5;
    const int lane = threadIdx.x & 31;
    const int b    = blockIdx.x * 8 + wid;
    const int half = lane >> 4;                // 0: K=d0,d1  1: K=d2,d3
    const int m    = lane & 15;                // A-matrix row / C column

    const int uu = u[b];
    const int vv = v[b];
    const float4* urow = (const float4*)(z + (size_t)uu * DD);
    const float4* vrow = (const float4*)(z + (size_t)vv * DD);
    float4 u4 = urow[lane];
    float4 v4 = vrow[lane];
    *(float4*)&su[wid][lane * 4] = u4;
    __syncthreads();

    float pos     = wave_sum(dot4(u4, v4));                // all lanes
    float regpart = dot4(u4, u4) + dot4(v4, v4);           // per-lane chunk of |u|^2+|v|^2

    float ww = w[b];
    float sgn = (ww > 0.f) ? 1.f : ((ww < 0.f) ? -1.f : 0.f);
    float a_coef = (mode == 0) ? (2.f - sgn) : (2.f + sgn);

    float sls = 0.f;                                       // sum log_sigmoid (lanes m==0 only)

    #pragma unroll
    for (int t = 0; t < 2; ++t) {                          // two 16-row k tiles
        const float* nrow = z + (size_t)n[b * KK + t * 16 + m] * DD;
        v8f c = {0.f, 0.f, 0.f, 0.f, 0.f, 0.f, 0.f, 0.f};
        // partial unroll: keeps ~4 K-steps of loads in flight without VGPR spills
        #pragma unroll 4
        for (int s = 0; s < 32; ++s) {                     // K-steps of 4 dims
            int d0 = 4 * s + 2 * half;
            v2f A;  A.x = nrow[d0];          A.y = nrow[d0 + 1];
            v2f Bv; Bv.x = su[wid][d0];      Bv.y = su[wid][d0 + 1];
            regpart += A.x * A.x + A.y * A.y;              // |n_k|^2 pieces (disjoint per lane)
            c = __builtin_amdgcn_wmma_f32_16x16x4_f32(
                    false, A, false, Bv, (short)0, c, false, false);
        }
        // C layout: lanes 0-15 col N=lane rows M=r; lanes 16-31 rows M=8+r.
        // All 16 columns identical (u broadcast) -> read column N=0 (lanes 0 and 16).
        if (m == 0) {
            #pragma unroll
            for (int r = 0; r < 8; ++r) {
                float neg = c[r];
                float arg = (mode == 0) ? (a_coef * pos - neg) : (neg - a_coef * pos);
                sls += log_sigmoid(arg);
            }
        }
    }

    float tot_reg = wave_sum(regpart);
    float tot_sls = wave_sum(sls);
    if (lane == 0)
        atomicAdd(out, -tot_sls + REGC * tot_reg);
}

// Contrastive term for a (z1, z2) view pair. One wave per b; lane owns 4 dims.
__global__ void k_contrast(const float* __restrict__ z1, const float* __restrict__ z2,
                           const int* __restrict__ u, const int* __restrict__ n,
                           float* out) {
    const int wid  = threadIdx.x >> 5;
    const int lane = threadIdx.x & 31;
    const int b    = blockIdx.x * 8 + wid;

    const int uu = u[b];
    float4 a1 = ((const float4*)(z1 + (size_t)uu * DD))[lane];
    float4 a2 = ((const float4*)(z2 + (size_t)uu * DD))[lane];

    float inv1 = 1.f / fmaxf(sqrtf(wave_sum(dot4(a1, a1))), 1e-12f);
    float inv2 = 1.f / fmaxf(sqrtf(wave_sum(dot4(a2, a2))), 1e-12f);
    float4 u3, u4;
    u3.x = a1.x * inv1; u3.y = a1.y * inv1; u3.z = a1.z * inv1; u3.w = a1.w * inv1;
    u4.x = a2.x * inv2; u4.y = a2.y * inv2; u4.z = a2.z * inv2; u4.w = a2.w * inv2;

    // per-dim norms over the K axis; keep row k=0
    float4 s1 = {0, 0, 0, 0}, s2 = {0, 0, 0, 0}, r1 = {0, 0, 0, 0}, r2 = {0, 0, 0, 0};
    for (int k = 0; k < KK; ++k) {
        int nk = n[b * KK + k];
        float4 t1 = ((const float4*)(z1 + (size_t)nk * DD))[lane];
        float4 t2 = ((const float4*)(z2 + (size_t)nk * DD))[lane];
        s1.x += t1.x * t1.x; s1.y += t1.y * t1.y; s1.z += t1.z * t1.z; s1.w += t1.w * t1.w;
        s2.x += t2.x * t2.x; s2.y += t2.y * t2.y; s2.z += t2.z * t2.z; s2.w += t2.w * t2.w;
        if (k == 0) { r1 = t1; r2 = t2; }
    }
    float4 n3, n4;
    n3.x = r1.x / fmaxf(sqrtf(s1.x), 1e-12f); n3.y = r1.y / fmaxf(sqrtf(s1.y), 1e-12f);
    n3.z = r1.z / fmaxf(sqrtf(s1.z), 1e-12f); n3.w = r1.w / fmaxf(sqrtf(s1.w), 1e-12f);
    n4.x = r2.x / fmaxf(sqrtf(s2.x), 1e-12f); n4.y = r2.y / fmaxf(sqrtf(s2.y), 1e-12f);
    n4.z = r2.z / fmaxf(sqrtf(s2.z), 1e-12f); n4.w = r2.w / fmaxf(sqrtf(s2.w), 1e-12f);

    float duu = wave_sum(dot4(u3, u4));
    float d13 = wave_sum(dot4(u3, n3));
    float d24 = wave_sum(dot4(u4, n4));

    if (lane == 0) {
        float lp = duu / TAU;
        float p  = expf(lp);
        float ng = expf(d13 / TAU) + expf(d24 / TAU);
        atomicAdd(out, logf(p + ng) - lp);   // == -log(p / (p + ng))
    }
}

// ---------------- launch ----------------

extern "C" void kernel_launch(void* const* d_in, const int* in_sizes, int n_in,
                              void* d_out, int out_size, void* d_ws, size_t ws_size,
                              hipStream_t stream) {
    const int*   u      = (const int*)d_in[0];
    const int*   v      = (const int*)d_in[1];
    const int*   n      = (const int*)d_in[2];
    const float* w      = (const float*)d_in[3];
    const int*   ep     = (const int*)d_in[4];
    const int*   en     = (const int*)d_in[5];
    const int*   ep1    = (const int*)d_in[6];
    const int*   ep2    = (const int*)d_in[7];
    const int*   en1    = (const int*)d_in[8];
    const int*   en2    = (const int*)d_in[9];
    const float* Epos   = (const float*)d_in[10];
    const float* Eneg   = (const float*)d_in[11];
    const float* Eitem  = (const float*)d_in[12];
    const float* Eitemn = (const float*)d_in[13];
    float* out = (float*)d_out;

    float* A = (float*)d_ws;
    float* B = A + NB_ELEMS;
    float* C = B + NB_ELEMS;

    const int ELEMS_BLOCKS = (int)((NB_VEC4 + 255) / 256);   // 18750
    const int EDGE_BLOCKS  = EE / 2;                         // 1,000,000
    const int BPR_BLOCKS   = BB / 8;                         // 1024

    // z = (x0 + conv(x0) + conv^2(x0)) / 3 ; result lands in x1
    auto prop = [&](const float* Eu, const float* Ei, const int* ei,
                    float* x1, float* x2) {
        k_concat_copy  <<<ELEMS_BLOCKS, 256, 0, stream>>>((float4*)x1, (const float4*)Eu, (const float4*)Ei);
        k_scatter_concat<<<EDGE_BLOCKS, 256, 0, stream>>>(x1, ei, Eu, Ei);
        k_copy         <<<ELEMS_BLOCKS, 256, 0, stream>>>((float4*)x2, (const float4*)x1);
        k_scatter      <<<EDGE_BLOCKS, 256, 0, stream>>>(x2, ei, x1);
        k_finalize     <<<ELEMS_BLOCKS, 256, 0, stream>>>((float4*)x1, (const float4*)x2,
                                                          (const float4*)Eu, (const float4*)Ei);
    };

    k_zero<<<1, 1, 0, stream>>>(out);

    // signed BPR, positive graph
    prop(Epos, Eitem, ep, A, B);
    k_bpr<<<BPR_BLOCKS, 256, 0, stream>>>(A, u, v, n, w, out, 0);

    // signed BPR, negative graph
    prop(Eneg, Eitemn, en, A, B);
    k_bpr<<<BPR_BLOCKS, 256, 0, stream>>>(A, u, v, n, w, out, 1);

    // contrastive, positive views
    prop(Epos, Eitem, ep1, A, B);       // zp1 -> A
    prop(Epos, Eitem, ep2, B, C);       // zp2 -> B
    k_contrast<<<BPR_BLOCKS, 256, 0, stream>>>(A, B, u, n, out);

    // contrastive, negative views
    prop(Eneg, Eitemn, en1, A, C);      // zn1 -> A
    prop(Eneg, Eitemn, en2, B, C);      // zn2 -> B
    k_contrast<<<BPR_BLOCKS, 256, 0, stream>>>(A, B, u, n, out);
}